// CPEmbedding_16587163697700
// MI455X (gfx1250) — compile-verified
//
#include <hip/hip_runtime.h>

// CP tensorized embedding, fused: decompose index -> Hadamard of U-factor rows
// -> 16x32 @ 32x128 GEMM via V_WMMA_F32_16X16X4_F32 -> LDS-staged, fully
// coalesced 512B-per-instruction row stores. Never materializes the table.

typedef __attribute__((ext_vector_type(2))) float v2f;
typedef __attribute__((ext_vector_type(8))) float v8f;

#define RANK 32
#define EMB  128
#define WPB  8          // waves per block (256 threads, wave32)

// LDS (floats): sU0 3200 | sU1 3200 | sU2 1600 | sB 4096  = 48384 B tables
//               sOut 8 waves * 16*128                      = 65536 B staging
// total ~114 KB / workgroup (CDNA5 allows up to 320 KB per WGP)

__global__ __launch_bounds__(256) void cp_embed_wmma_kernel(
    const int*   __restrict__ x,
    const float* __restrict__ U0,
    const float* __restrict__ U1,
    const float* __restrict__ U2,
    const float* __restrict__ V0,
    const float* __restrict__ V1,
    float*       __restrict__ out,
    int n_tokens)
{
    __shared__ float sTab[12096];
    __shared__ float sOut[WPB][16 * EMB];

    float* sU0 = sTab;
    float* sU1 = sTab + 3200;
    float* sU2 = sTab + 6400;
    float* sB  = sTab + 8000;

    const int tid = threadIdx.x;

    // Stage factor matrices + emb-side Khatri-Rao table into LDS.
    for (int i = tid; i < 3200; i += 256) sU0[i] = U0[i];
    for (int i = tid; i < 3200; i += 256) sU1[i] = U1[i];
    for (int i = tid; i < 1600; i += 256) sU2[i] = U2[i];
    for (int i = tid; i < 128 * 32; i += 256) {
        int r  = i & 31;
        int de = i >> 5;
        sB[i] = V0[(de >> 4) * 32 + r] * V1[(de & 15) * 32 + r];
    }
    __syncthreads();

    const int wave = tid >> 5;
    const int lane = tid & 31;
    const int half = lane >> 4;      // K-pair select / output row half
    const int lm   = lane & 15;

    float*    stg        = sOut[wave];
    const int nGroups    = (n_tokens + 15) >> 4;
    const int waveStride = gridDim.x * WPB;

    // Grid-stride over 16-token groups (wave-uniform: EXEC all-1s for WMMA).
    for (int g = blockIdx.x * WPB + wave; g < nGroups; g += waveStride) {
        const int tbase = g * 16;

        // ---- A fragment: w[token][r] = U0[a][r]*U1[b][r]*U2[c][r] ----
        int t   = tbase + lm;
        int tt  = (t < n_tokens) ? t : (n_tokens - 1);   // clamp tail reads
        int idx = x[tt];
        int a_i = idx / 5000;                            // idx = (a*100+b)*50+c
        int rem = idx - a_i * 5000;
        int b_i = rem / 50;
        int c_i = rem - b_i * 50;

        const float* u0 = sU0 + a_i * RANK;
        const float* u1 = sU1 + b_i * RANK;
        const float* u2 = sU2 + c_i * RANK;

        v2f afrag[8];
        #pragma unroll
        for (int s = 0; s < 8; ++s) {
            int r0 = 4 * s + 2 * half;
            afrag[s].x = u0[r0]     * u1[r0]     * u2[r0];
            afrag[s].y = u0[r0 + 1] * u1[r0 + 1] * u2[r0 + 1];
        }

        // ---- 8 N-tiles of 16 emb cols; K=32 as 8 x (K=4) f32 WMMAs ----
        const float* bcol = sB + lm * RANK;
        #pragma unroll
        for (int j = 0; j < 8; ++j) {
            const float* bj = bcol + (16 * j) * RANK;
            v8f acc = {};
            #pragma unroll
            for (int s = 0; s < 8; ++s) {
                int k0 = 4 * s + 2 * half;
                v2f bfrag;
                bfrag.x = bj[k0];
                bfrag.y = bj[k0 + 1];
                acc = __builtin_amdgcn_wmma_f32_16x16x4_f32(
                    false, afrag[s], false, bfrag, (short)0, acc, false, false);
            }
            // D: VGPR i -> row (i + 8*half), col lm of tile j -> LDS stage
            #pragma unroll
            for (int i = 0; i < 8; ++i)
                stg[(i + 8 * half) * EMB + 16 * j + lm] = acc[i];
        }

        // ---- Flush 16x128 tile: one 512B contiguous store per row ----
        // (same-wave DS ops are in-order; no barrier needed)
        int nvalid = n_tokens - tbase;
        if (nvalid > 16) nvalid = 16;
        const int c4 = lane * 4;
        for (int i = 0; i < nvalid; ++i) {               // wave-uniform bound
            const float4 v = *(const float4*)(stg + i * EMB + c4);
            *(float4*)(out + (size_t)(tbase + i) * EMB + c4) = v;
        }
    }
}

extern "C" void kernel_launch(void* const* d_in, const int* in_sizes, int n_in,
                              void* d_out, int out_size, void* d_ws, size_t ws_size,
                              hipStream_t stream) {
    const int*   x  = (const int*)  d_in[0];
    const float* U0 = (const float*)d_in[1];
    const float* U1 = (const float*)d_in[2];
    const float* U2 = (const float*)d_in[3];
    const float* V0 = (const float*)d_in[4];
    const float* V1 = (const float*)d_in[5];
    float* out = (float*)d_out;

    int n_tokens = in_sizes[0];                    // 1024*200 = 204800
    int nGroups  = (n_tokens + 15) / 16;           // 16-token wave groups
    int blocks   = (nGroups + WPB - 1) / WPB;
    if (blocks > 640) blocks = 640;                // grid-stride amortizes staging

    cp_embed_wmma_kernel<<<blocks, 256, 0, stream>>>(x, U0, U1, U2, V0, V1, out, n_tokens);
}